// SwitchGate_89824946028711
// MI455X (gfx1250) — compile-verified
//
#include <hip/hip_runtime.h>

typedef float v2f __attribute__((ext_vector_type(2)));
typedef float v8f __attribute__((ext_vector_type(8)));

#define DIM_K   2048
#define NEXP    64
#define MTILE   128      // tokens per workgroup (8 waves x 16 tokens)
#define KTILE   64       // K staged per LDS stage
#define XPITCH  68       // x tile row pitch (floats): conflict-free A-frag reads
#define WPITCH  130      // w tile pair-row pitch (floats): 64 experts * 2 + 2 pad

// ---------------------------------------------------------------------------
// K0: zero the per-expert denominator accumulator
// ---------------------------------------------------------------------------
__global__ void zero_denom_kernel(float* __restrict__ denom) {
    if (threadIdx.x < NEXP) denom[threadIdx.x] = 0.0f;
}

// ---------------------------------------------------------------------------
// K1: logits GEMM (WMMA f32 16x16x4) + bias + streamed softmax-top1.
//     Software-pipelined: next K-stage's global loads are issued before the
//     current stage's WMMA work so HBM latency hides under compute.
//     w tile is stored pair-interleaved so each B fragment is ONE ds_load_b64
//     landing directly in an even-aligned VGPR pair (no v_mov repacking).
// ---------------------------------------------------------------------------
__global__ __launch_bounds__(256) void gate_gemm_top1_kernel(
    const float* __restrict__ x,        // [tokens, DIM_K]
    const float* __restrict__ w,        // [NEXP, DIM_K]
    const float* __restrict__ bg,       // [NEXP]
    int*   __restrict__ top1_idx,       // [tokens]
    float* __restrict__ top1_val,       // [tokens]
    float* __restrict__ denom)          // [NEXP]
{
    __shared__ float xs[MTILE * XPITCH];           // 34816 B: x tile [row][k]
    __shared__ float wsh[(KTILE / 2) * WPITCH];    // 16640 B: w tile [k/2][2e+(k&1)]

    const int tid  = threadIdx.x;
    const int wave = tid >> 5;
    const int lane = tid & 31;
    const int half = lane >> 4;
    const int ln   = lane & 15;
    const int t0   = blockIdx.x * MTILE;

    v8f acc0 = {}, acc1 = {}, acc2 = {}, acc3 = {};

    float4 xr[8];   // staged x tile: 8 float4/thread  (128 rows x 64 k)
    float4 wr[4];   // staged w tile: 4 float4/thread  (64 experts x 64 k)

    // ---- prologue: issue stage-0 global loads into registers
    #pragma unroll
    for (int j = 0; j < 8; ++j) {
        const int idx = tid + j * 256, row = idx >> 4, col = (idx & 15) << 2;
        xr[j] = *reinterpret_cast<const float4*>(&x[(size_t)(t0 + row) * DIM_K + col]);
    }
    #pragma unroll
    for (int j = 0; j < 4; ++j) {
        const int idx = tid + j * 256, e = idx >> 4, k = (idx & 15) << 2;
        wr[j] = *reinterpret_cast<const float4*>(&w[(size_t)e * DIM_K + k]);
    }

    for (int kb = 0; kb < DIM_K; kb += KTILE) {
        // ---- drain staged registers into LDS
        #pragma unroll
        for (int j = 0; j < 8; ++j) {
            const int idx = tid + j * 256, row = idx >> 4, col = (idx & 15) << 2;
            float* d = &xs[row * XPITCH + col];
            d[0] = xr[j].x; d[1] = xr[j].y; d[2] = xr[j].z; d[3] = xr[j].w;
        }
        #pragma unroll
        for (int j = 0; j < 4; ++j) {
            const int idx = tid + j * 256, e = idx >> 4, k = (idx & 15) << 2;
            *reinterpret_cast<v2f*>(&wsh[(k >> 1)       * WPITCH + 2 * e]) = (v2f){wr[j].x, wr[j].y};
            *reinterpret_cast<v2f*>(&wsh[((k >> 1) + 1) * WPITCH + 2 * e]) = (v2f){wr[j].z, wr[j].w};
        }
        __syncthreads();

        // ---- issue NEXT stage's global loads now; they complete under compute
        const int kn = kb + KTILE;
        if (kn < DIM_K) {
            #pragma unroll
            for (int j = 0; j < 8; ++j) {
                const int idx = tid + j * 256, row = idx >> 4, col = (idx & 15) << 2;
                xr[j] = *reinterpret_cast<const float4*>(&x[(size_t)(t0 + row) * DIM_K + kn + col]);
            }
            #pragma unroll
            for (int j = 0; j < 4; ++j) {
                const int idx = tid + j * 256, e = idx >> 4, k = (idx & 15) << 2;
                wr[j] = *reinterpret_cast<const float4*>(&w[(size_t)e * DIM_K + kn + k]);
            }
            // nudge the stage-after-next x tile toward L2
            if (kn + KTILE < DIM_K) {
                const int row = tid >> 1, part = (tid & 1) << 5;
                __builtin_prefetch(&x[(size_t)(t0 + row) * DIM_K + kn + KTILE + part], 0, 0);
            }
        }

        // ---- compute: wave owns tokens [t0+wave*16, +16), all 64 experts
        // A frag (16x4 f32): lane = M, VGPR v at K = ks + 2*half + v
        // B frag (4x16 f32): VGPR v at K = ks + 2*half + v, lane%16 = N
        const float* xrow = &xs[(wave * 16 + ln) * XPITCH + 2 * half];
        #pragma unroll
        for (int ks = 0; ks < KTILE; ks += 4) {
            const v2f a = *reinterpret_cast<const v2f*>(&xrow[ks]);
            const float* wrow = &wsh[((ks >> 1) + half) * WPITCH + 2 * ln];
            const v2f b0 = *reinterpret_cast<const v2f*>(&wrow[ 0]);
            const v2f b1 = *reinterpret_cast<const v2f*>(&wrow[32]);
            const v2f b2 = *reinterpret_cast<const v2f*>(&wrow[64]);
            const v2f b3 = *reinterpret_cast<const v2f*>(&wrow[96]);
            acc0 = __builtin_amdgcn_wmma_f32_16x16x4_f32(false, a, false, b0,
                                                         (short)0, acc0, false, false);
            acc1 = __builtin_amdgcn_wmma_f32_16x16x4_f32(false, a, false, b1,
                                                         (short)0, acc1, false, false);
            acc2 = __builtin_amdgcn_wmma_f32_16x16x4_f32(false, a, false, b2,
                                                         (short)0, acc2, false, false);
            acc3 = __builtin_amdgcn_wmma_f32_16x16x4_f32(false, a, false, b3,
                                                         (short)0, acc3, false, false);
        }
        __syncthreads();
    }

    // ---- add bias (expert index depends only on N-tile and ln)
    acc0 += bg[ 0 + ln];
    acc1 += bg[16 + ln];
    acc2 += bg[32 + ln];
    acc3 += bg[48 + ln];

    // ---- per-token streamed softmax-top1 across 4 tiles x 16 lanes/half
    // C layout: VGPR r, half h -> token row r + 8h; lane%16 -> expert tile col
    const int wtok0 = t0 + wave * 16;
    #pragma unroll
    for (int r = 0; r < 8; ++r) {
        const float l0 = acc0[r], l1 = acc1[r], l2 = acc2[r], l3 = acc3[r];
        float m = fmaxf(fmaxf(l0, l1), fmaxf(l2, l3));
        #pragma unroll
        for (int s = 1; s < 16; s <<= 1) m = fmaxf(m, __shfl_xor(m, s, 32));
        // sum of exp(l - m); includes exp(0)=1 at the max -> top1 score = 1/se
        float se = __expf(l0 - m) + __expf(l1 - m) + __expf(l2 - m) + __expf(l3 - m);
        #pragma unroll
        for (int s = 1; s < 16; s <<= 1) se += __shfl_xor(se, s, 32);
        // first-max expert index (jnp.argmax tie-break: lowest index)
        int cand = 1 << 30;
        if (l3 == m) cand = 48 + ln;
        if (l2 == m) cand = 32 + ln;
        if (l1 == m) cand = 16 + ln;
        if (l0 == m) cand =  0 + ln;
        #pragma unroll
        for (int s = 1; s < 16; s <<= 1) cand = min(cand, __shfl_xor(cand, s, 32));

        if (ln == 0) {
            const int   token = wtok0 + r + 8 * half;
            const float score = 1.0f / se;
            top1_idx[token] = cand;
            top1_val[token] = score;
            atomicAdd(&denom[cand], score);
        }
    }
}

// ---------------------------------------------------------------------------
// K2: scatter + normalize: out[t][e] = (e==idx) ? score*cap/(denom+eps) : 0
// ---------------------------------------------------------------------------
__global__ __launch_bounds__(256) void scatter_norm_kernel(
    const int*   __restrict__ top1_idx,
    const float* __restrict__ top1_val,
    const float* __restrict__ denom,
    float* __restrict__ out,
    int tokens, float capacity)
{
    const int gid = blockIdx.x * 256 + threadIdx.x;   // one per (token, 4-expert group)
    const int t  = gid >> 4;
    const int e0 = (gid & 15) << 2;
    if (t >= tokens) return;
    const int idx = top1_idx[t];
    float4 v = make_float4(0.0f, 0.0f, 0.0f, 0.0f);
    if (idx >= e0 && idx < e0 + 4) {
        const float val = top1_val[t] * capacity / (denom[idx] + 1e-6f);
        reinterpret_cast<float*>(&v)[idx - e0] = val;
    }
    *reinterpret_cast<float4*>(&out[(size_t)t * NEXP + e0]) = v;
}

// ---------------------------------------------------------------------------
extern "C" void kernel_launch(void* const* d_in, const int* in_sizes, int n_in,
                              void* d_out, int out_size, void* d_ws, size_t ws_size,
                              hipStream_t stream) {
    const float* x  = (const float*)d_in[0];   // [8,4096,2048] f32
    const float* w  = (const float*)d_in[1];   // [64,2048]     f32
    const float* bg = (const float*)d_in[2];   // [64]          f32
    float* out = (float*)d_out;                // [8,4096,64]   f32

    const int tokens = in_sizes[0] / DIM_K;    // 32768
    const float capacity = (float)tokens;      // CAPACITY_FACTOR = 1.0

    char* ws = (char*)d_ws;
    int*   top1_idx = (int*)ws;
    float* top1_val = (float*)(ws + (size_t)tokens * sizeof(int));
    float* denom    = (float*)(ws + (size_t)tokens * (sizeof(int) + sizeof(float)));

    zero_denom_kernel<<<1, 64, 0, stream>>>(denom);
    gate_gemm_top1_kernel<<<tokens / MTILE, 256, 0, stream>>>(
        x, w, bg, top1_idx, top1_val, denom);
    scatter_norm_kernel<<<(tokens * 16 + 255) / 256, 256, 0, stream>>>(
        top1_idx, top1_val, denom, out, tokens, capacity);
}